// LGESQL_76209899700247
// MI455X (gfx1250) — compile-verified
//
#include <hip/hip_runtime.h>
#include <math.h>

#define E_NODES 20000
#define LE_EDGES 320000
#define NDIM_ 256
#define HEADS 8
#define DK_ 32

typedef __attribute__((ext_vector_type(2))) float v2f;
typedef __attribute__((ext_vector_type(8))) float v8f;

// ---------------------------------------------------------------------------
// fp32 GEMM: C[M,N] = A[M,K] @ W[K,N] (+bias[N]) (+addm[M,N]) (relu?)
// Compile-time K/N -> immediate-offset global loads.
// One wave computes a 16x64 tile: 4 x V_WMMA_F32_16X16X4_F32 accumulators,
// A fragment reused 4x. Grid must exactly cover (M/16)*(N/64) waves.
// ---------------------------------------------------------------------------
template <int K, int N, bool HAS_BIAS, bool HAS_ADD, bool RELU>
__global__ __launch_bounds__(256) void lgesql_gemm_wmma_f32(
    const float* __restrict__ A, const float* __restrict__ W,
    const float* __restrict__ bias, const float* __restrict__ addm,
    float* __restrict__ C)
{
  constexpr int TN = 64;                 // cols per wave
  constexpr int tilesN = N / TN;
  const int wid  = (blockIdx.x * blockDim.x + threadIdx.x) >> 5;
  const int lane = threadIdx.x & 31;
  const int tm = wid / tilesN;
  const int tn = wid - tm * tilesN;
  const int half = lane >> 4;            // 0: K {0,1}, 1: K {2,3}
  const int l    = lane & 15;

  const float* Ap = A + (size_t)(tm * 16 + l) * K + 2 * half;
  const float* Bp = W + tn * TN + l + (size_t)(2 * half) * N;

  v8f acc[4] = {};
  #pragma unroll 4
  for (int k0 = 0; k0 < K; k0 += 4) {
    v2f a = *(const v2f*)(Ap + k0);                 // A 16x4 fragment (b64)
    const float* b = Bp + (size_t)k0 * N;
    v2f b0, b1, b2, b3;                             // B 4x16 fragments
    b0.x = b[0];   b0.y = b[N];
    b1.x = b[16];  b1.y = b[16 + N];
    b2.x = b[32];  b2.y = b[32 + N];
    b3.x = b[48];  b3.y = b[48 + N];
    acc[0] = __builtin_amdgcn_wmma_f32_16x16x4_f32(false, a, false, b0, (short)0, acc[0], false, false);
    acc[1] = __builtin_amdgcn_wmma_f32_16x16x4_f32(false, a, false, b1, (short)0, acc[1], false, false);
    acc[2] = __builtin_amdgcn_wmma_f32_16x16x4_f32(false, a, false, b2, (short)0, acc[2], false, false);
    acc[3] = __builtin_amdgcn_wmma_f32_16x16x4_f32(false, a, false, b3, (short)0, acc[3], false, false);
  }

  // D layout: VGPR r -> lanes 0-15: M=r, N=l ; lanes 16-31: M=r+8, N=l
  #pragma unroll
  for (int j = 0; j < 4; ++j) {
    const int col = tn * TN + j * 16 + l;
    const float bv = HAS_BIAS ? bias[col] : 0.0f;
    #pragma unroll
    for (int r = 0; r < 8; ++r) {
      const int row = tm * 16 + r + 8 * half;
      float v = acc[j][r] + bv;
      if (HAS_ADD) v += addm[(size_t)row * N + col];
      if (RELU)    v = fmaxf(v, 0.0f);
      C[(size_t)row * N + col] = v;
    }
  }
}

// ---------------------------------------------------------------------------
// Edge attention: one wave per line-graph edge.
// score[e,h] = exp(clip(dot(k[src,h,:], q[dst,h,:]) / sqrt(DK), -5, 5))
// wv[dst] += v[src] * score ; z[dst,h] += score   (fp32 global atomics)
// ---------------------------------------------------------------------------
__global__ __launch_bounds__(256) void lgesql_score_scatter(
    const float* __restrict__ q, const float* __restrict__ k,
    const float* __restrict__ v, const int* __restrict__ lg_src,
    const int* __restrict__ lg_dst, float* __restrict__ wv,
    float* __restrict__ z)
{
  const int e    = (blockIdx.x * blockDim.x + threadIdx.x) >> 5;
  const int lane = threadIdx.x & 31;
  const int s = lg_src[e];
  const int d = lg_dst[e];
  const float* kr = k + (size_t)s * NDIM_;
  const float* qr = q + (size_t)d * NDIM_;
  const float* vr = v + (size_t)s * NDIM_;
  float* wvr = wv + (size_t)d * NDIM_;

  #pragma unroll
  for (int h = 0; h < HEADS; ++h) {
    const int c = h * DK_ + lane;
    float p = kr[c] * qr[c];
    #pragma unroll
    for (int off = 16; off > 0; off >>= 1)
      p += __shfl_xor(p, off, 32);
    float sc = p * 0.17677669529663687f;          // 1/sqrt(32)
    sc = fminf(5.0f, fmaxf(-5.0f, sc));
    sc = expf(sc);
    atomicAdd(&wvr[c], vr[c] * sc);
    if (lane == 0) atomicAdd(&z[(size_t)d * HEADS + h], sc);
  }
}

// o = wv / z (per-head broadcast), in place over wv
__global__ __launch_bounds__(256) void lgesql_normalize(
    float* __restrict__ wv, const float* __restrict__ z)
{
  const size_t i = (size_t)blockIdx.x * blockDim.x + threadIdx.x;
  const size_t row = i >> 8;            // / NDIM_
  const size_t col = i & 255;
  wv[i] = wv[i] / z[row * HEADS + (col >> 5)];
}

// LayerNorm over last dim (256), one wave per row
__global__ __launch_bounds__(256) void lgesql_layernorm(
    const float* __restrict__ in, const float* __restrict__ g,
    const float* __restrict__ b, float* __restrict__ out)
{
  const int row  = (blockIdx.x * blockDim.x + threadIdx.x) >> 5;
  const int lane = threadIdx.x & 31;
  const float* r = in + (size_t)row * NDIM_;
  float vals[8];
  float sum = 0.0f, sq = 0.0f;
  #pragma unroll
  for (int t = 0; t < 8; ++t) {
    float x = r[lane + 32 * t];
    vals[t] = x; sum += x; sq += x * x;
  }
  #pragma unroll
  for (int off = 16; off > 0; off >>= 1) {
    sum += __shfl_xor(sum, off, 32);
    sq  += __shfl_xor(sq,  off, 32);
  }
  const float mu   = sum * (1.0f / NDIM_);
  const float var  = sq * (1.0f / NDIM_) - mu * mu;
  const float rstd = rsqrtf(var + 1e-5f);
  float* o = out + (size_t)row * NDIM_;
  #pragma unroll
  for (int t = 0; t < 8; ++t) {
    const int c = lane + 32 * t;
    o[c] = (vals[t] - mu) * rstd * g[c] + b[c];
  }
}

// ---------------------------------------------------------------------------
extern "C" void kernel_launch(void* const* d_in, const int* in_sizes, int n_in,
                              void* d_out, int out_size, void* d_ws, size_t ws_size,
                              hipStream_t stream) {
  const float* x     = (const float*)d_in[0];
  const float* src_x = (const float*)d_in[1];
  const float* dst_x = (const float*)d_in[2];
  const float* Wq    = (const float*)d_in[3];
  const float* bq    = (const float*)d_in[4];
  const float* Wk    = (const float*)d_in[5];
  const float* Wv    = (const float*)d_in[6];
  const float* Wo    = (const float*)d_in[7];
  const float* bo    = (const float*)d_in[8];
  const float* ln1_g = (const float*)d_in[9];
  const float* ln1_b = (const float*)d_in[10];
  const float* W1    = (const float*)d_in[11];
  const float* b1    = (const float*)d_in[12];
  const float* W2    = (const float*)d_in[13];
  const float* b2    = (const float*)d_in[14];
  const float* ln2_g = (const float*)d_in[15];
  const float* ln2_b = (const float*)d_in[16];
  const int*   lg_src = (const int*)d_in[17];
  const int*   lg_dst = (const int*)d_in[18];
  float* out = (float*)d_out;

  const size_t EN = (size_t)E_NODES * NDIM_;       // 5,120,000 floats
  float* ws = (float*)d_ws;
  float* qf   = ws;                 // q ; later reused as t1 (pre-LN1)
  float* kf   = ws + EN;            // k ; later reused as h
  float* vf   = ws + 2 * EN;        // v ; later reused as t2 (pre-LN2)
  float* wvf  = ws + 3 * EN;        // wv ; later o (in place)
  float* zf   = ws + 4 * EN;        // z [E, H]
  float* ffn1 = ws + 4 * EN + (size_t)E_NODES * HEADS;  // [E, 1024]

  // zero the segment-sum accumulators (wv and z are contiguous)
  hipMemsetAsync(wvf, 0, (EN + (size_t)E_NODES * HEADS) * sizeof(float), stream);

  const dim3 blk(256);
  // waves = (M/16)*(N/64), 8 waves per block — all divide exactly
  const int g256  = (E_NODES / 16) * (NDIM_ / 64) / 8;        // 625 blocks
  const int g1024 = (E_NODES / 16) * (4 * NDIM_ / 64) / 8;    // 2500 blocks

  // q = x@Wq + bq + src_x ; k = x@Wk ; v = x@Wv + dst_x
  lgesql_gemm_wmma_f32<256, 256, true,  true,  false>
      <<<g256, blk, 0, stream>>>(x, Wq, bq, src_x, qf);
  lgesql_gemm_wmma_f32<256, 256, false, false, false>
      <<<g256, blk, 0, stream>>>(x, Wk, nullptr, nullptr, kf);
  lgesql_gemm_wmma_f32<256, 256, false, true,  false>
      <<<g256, blk, 0, stream>>>(x, Wv, nullptr, dst_x, vf);

  // edge attention scatter: one wave per edge, 8 waves/block
  lgesql_score_scatter<<<LE_EDGES / 8, blk, 0, stream>>>(qf, kf, vf,
                                                         lg_src, lg_dst, wvf, zf);

  // o = wv / z
  lgesql_normalize<<<(int)(EN / 256), blk, 0, stream>>>(wvf, zf);

  // t1 = x + o@Wo + bo  (into q buffer)
  lgesql_gemm_wmma_f32<256, 256, true,  true,  false>
      <<<g256, blk, 0, stream>>>(wvf, Wo, bo, x, qf);
  // h = LN1(t1) (into k buffer)
  lgesql_layernorm<<<E_NODES / 8, blk, 0, stream>>>(qf, ln1_g, ln1_b, kf);

  // ffn1 = relu(h@W1 + b1)  [E, 1024]
  lgesql_gemm_wmma_f32<256, 1024, true, false, true>
      <<<g1024, blk, 0, stream>>>(kf, W1, b1, nullptr, ffn1);
  // t2 = h + ffn1@W2 + b2  (into v buffer)
  lgesql_gemm_wmma_f32<1024, 256, true, true, false>
      <<<g256, blk, 0, stream>>>(ffn1, W2, b2, kf, vf);
  // out = LN2(t2)
  lgesql_layernorm<<<E_NODES / 8, blk, 0, stream>>>(vf, ln2_g, ln2_b, out);
}